// BERT_BERT_MLP_50861002719745
// MI455X (gfx1250) — compile-verified
//
#include <hip/hip_runtime.h>
#include <hip/hip_bf16.h>

typedef __attribute__((ext_vector_type(2))) float v2f;
typedef __attribute__((ext_vector_type(8))) float v8f;

#define CRF_B     2048
#define CRF_T     512
#define CRF_OUT   23
#define CRF_K     25          // OUT + 2
#define CRF_START 23
#define CRF_STOP  24
#define NEG_BIG   (-1.0e30f)
#define LN2F      0.69314718055994531f

__device__ __forceinline__ float fast_log(float x) {
    // x guaranteed >= 1e-30 (normal range): raw v_log_f32 (log2) * ln2
    return LN2F * __builtin_amdgcn_logf(x);
}

// ---------------------------------------------------------------------------
// CRF forward scan in exp-space:
//   alpha'[j] = m + log( sum_i E[j][i] * exp(alpha[i]-m) ) + feat[j]
// E = exp(transitions) (25x25 zero-padded to 32x32) held in WMMA A-fragments.
// One wave (32 lanes) per block handles 16 batch columns for all T steps.
// alpha is kept in the WMMA f32 16x16 C/D register layout (two v8f tiles).
//
// Padded states (s >= 25) need no masking in the hot loop: E's padded rows
// and columns are zero, so padded alphas contribute nothing to the matvec
// and settle ~69 below the column max (log(1e-30) clamp). Only the final
// logsumexp masks them (stopT = NEG_BIG).
//
// The per-state update uses an inline-asm value barrier so the compiler
// cannot sink the load+log chain into per-state exec-guarded blocks; the
// sequence-length mask then reduces to a single v_cndmask per state and
// EXEC stays full for the entire loop.
// ---------------------------------------------------------------------------
__global__ __launch_bounds__(32) void crf_scan_kernel(
    const float* __restrict__ feats,      // [B, T, K]
    const int*   __restrict__ lengths,    // [B]
    const float* __restrict__ trans,      // [K, K]  (row = next state j, col = prev i)
    float*       __restrict__ fwd_out)    // [B]
{
    const int lane = threadIdx.x;     // 0..31 (wave32)
    const int c    = lane & 15;       // batch column within the 16-wide tile
    const int hi   = lane >> 4;       // which half of the wave
    const int b    = blockIdx.x * 16 + c;

    // ---- A fragments: E[j][i] = exp(trans[j][i]), zero-padded to 32x32.
    // 16x4 f32 A layout: lanes 0-15 hold M, lane half selects K offset +2.
    v2f Afrag[2][8];
#pragma unroll
    for (int mt = 0; mt < 2; ++mt) {
#pragma unroll
        for (int kc = 0; kc < 8; ++kc) {
            const int m  = mt * 16 + c;
            const int kb = kc * 4 + hi * 2;
            const bool v0 = (m < CRF_K) & (kb     < CRF_K);
            const bool v1 = (m < CRF_K) & (kb + 1 < CRF_K);
            const float t0 = trans[v0 ? (m * CRF_K + kb)     : 0];
            const float t1 = trans[v1 ? (m * CRF_K + kb + 1) : 0];
            v2f a;
            a.x = v0 ? __expf(t0) : 0.0f;
            a.y = v1 ? __expf(t1) : 0.0f;
            Afrag[mt][kc] = a;
        }
    }

    // ---- alpha in C/D layout: tile t, VGPR r  <->  state s = tile*16 + hi*8 + r
    float alpha[2][8];
    float stopT[2][8];
#pragma unroll
    for (int tile = 0; tile < 2; ++tile) {
#pragma unroll
        for (int r = 0; r < 8; ++r) {
            const int s = tile * 16 + hi * 8 + r;
            const bool real = (s < CRF_K);
            alpha[tile][r] = (s == CRF_START) ? 0.0f : NEG_BIG;
            const float tv = trans[CRF_STOP * CRF_K + (real ? s : 0)];
            stopT[tile][r] = real ? tv : NEG_BIG;
        }
    }

    const int len = lengths[b];
    const float* featb = feats + (size_t)b * CRF_T * CRF_K;

    // LDS staging tile for the B operand: vbuf[col][state], padded row (34)
    // keeps 8B alignment for f32x2 merges and breaks bank conflicts.
    __shared__ float vbuf[16][34];

    for (int t = 0; t < CRF_T; ++t) {
        // prefetch next step's feature row (clamped: no per-step branch)
        const int tn = (t + 1 < CRF_T) ? (t + 1) : t;
        __builtin_prefetch(featb + tn * CRF_K, 0, 0);   // global_prefetch_b8

        // ---- feature loads for this step: branchless clamped gather.
        float fvals[2][8];
#pragma unroll
        for (int tile = 0; tile < 2; ++tile)
#pragma unroll
            for (int r = 0; r < 8; ++r) {
                const int s  = tile * 16 + hi * 8 + r;
                const int sc = (s < CRF_K) ? s : 0;   // always in-bounds
                fvals[tile][r] = featb[t * CRF_K + sc];
            }

        // ---- per-column max (16 in-lane values + cross-half shuffle)
        float m = alpha[0][0];
#pragma unroll
        for (int tile = 0; tile < 2; ++tile)
#pragma unroll
            for (int r = 0; r < 8; ++r) m = fmaxf(m, alpha[tile][r]);
        m = fmaxf(m, __shfl_xor(m, 16, 32));

        // ---- v = exp(alpha - m), D-layout -> LDS (consecutive states/lane)
#pragma unroll
        for (int tile = 0; tile < 2; ++tile)
#pragma unroll
            for (int r = 0; r < 8; ++r)
                vbuf[c][tile * 16 + hi * 8 + r] = __expf(alpha[tile][r] - m);

        __syncthreads();

        // ---- rebuild B fragments (4x16 f32): lane holds rows kc*4+hi*2, +1
        v2f Bfrag[8];
#pragma unroll
        for (int kc = 0; kc < 8; ++kc) {
            const int row = kc * 4 + hi * 2;
            v2f bf; bf.x = vbuf[c][row]; bf.y = vbuf[c][row + 1];
            Bfrag[kc] = bf;
        }

        // ---- u = E * v via 16 chained V_WMMA_F32_16X16X4_F32
        v8f acc[2];
        acc[0] = {}; acc[1] = {};
#pragma unroll
        for (int kc = 0; kc < 8; ++kc) {
            acc[0] = __builtin_amdgcn_wmma_f32_16x16x4_f32(
                false, Afrag[0][kc], false, Bfrag[kc], (short)0, acc[0], false, false);
            acc[1] = __builtin_amdgcn_wmma_f32_16x16x4_f32(
                false, Afrag[1][kc], false, Bfrag[kc], (short)0, acc[1], false, false);
        }

        __syncthreads();

        // ---- alpha' = m + log(u) + feat ; value-barrier pins na in a VGPR
        // so the mask reduces to one v_cndmask per state (no exec churn).
        const bool live = (t < len);
#pragma unroll
        for (int tile = 0; tile < 2; ++tile) {
#pragma unroll
            for (int r = 0; r < 8; ++r) {
                const float u  = fmaxf(acc[tile][r], 1e-30f);
                float na = m + fast_log(u) + fvals[tile][r];
                asm volatile("" : "+v"(na));   // forbid sinking under exec mask
                alpha[tile][r] = live ? na : alpha[tile][r];
            }
        }
    }

    // ---- fwd[b] = logsumexp_s(alpha[s] + trans[STOP][s])
    float mm = -3.0e38f;
#pragma unroll
    for (int tile = 0; tile < 2; ++tile)
#pragma unroll
        for (int r = 0; r < 8; ++r)
            mm = fmaxf(mm, alpha[tile][r] + stopT[tile][r]);
    float sum = 0.0f;
#pragma unroll
    for (int tile = 0; tile < 2; ++tile)
#pragma unroll
        for (int r = 0; r < 8; ++r)
            sum += __expf((alpha[tile][r] + stopT[tile][r]) - mm);

    const float mo = __shfl_xor(mm, 16, 32);
    const float so = __shfl_xor(sum, 16, 32);
    const float M  = fmaxf(mm, mo);
    const float S  = __expf(mm - M) * sum + __expf(mo - M) * so;
    if (hi == 0) fwd_out[b] = M + fast_log(S);
}

// ---------------------------------------------------------------------------
// Gold path score: one thread per batch element.
// ---------------------------------------------------------------------------
__global__ void crf_gold_kernel(
    const float* __restrict__ feats, const int* __restrict__ tags,
    const int* __restrict__ lengths, const float* __restrict__ trans,
    float* __restrict__ gold_out)
{
    const int b = blockIdx.x * blockDim.x + threadIdx.x;
    if (b >= CRF_B) return;
    const int len = lengths[b];
    const float* fb = feats + (size_t)b * CRF_T * CRF_K;
    const int*   tb = tags  + (size_t)b * CRF_T;

    int   prev = CRF_START;
    int   last = CRF_START;
    float acc  = 0.0f;
    for (int t = 0; t < CRF_T; ++t) {
        const int  tag  = tb[t];
        const bool live = (t < len);
        float sc = trans[tag * CRF_K + prev] + fb[t * CRF_K + tag];
        asm volatile("" : "+v"(sc));
        acc  = live ? (acc + sc) : acc;
        prev = live ? tag : prev;
        last = live ? tag : last;
    }
    acc += trans[CRF_STOP * CRF_K + last];
    gold_out[b] = acc;
}

// ---------------------------------------------------------------------------
// mean(fwd - gold) -> out[0]; single block, deterministic tree reduction.
// ---------------------------------------------------------------------------
__global__ void crf_reduce_kernel(const float* __restrict__ fwd,
                                  const float* __restrict__ gold,
                                  float* __restrict__ out)
{
    __shared__ float red[256];
    float s = 0.0f;
    for (int b = threadIdx.x; b < CRF_B; b += 256) s += fwd[b] - gold[b];
    red[threadIdx.x] = s;
    __syncthreads();
    for (int off = 128; off > 0; off >>= 1) {
        if (threadIdx.x < off) red[threadIdx.x] += red[threadIdx.x + off];
        __syncthreads();
    }
    if (threadIdx.x == 0) out[0] = red[0] / (float)CRF_B;
}

extern "C" void kernel_launch(void* const* d_in, const int* in_sizes, int n_in,
                              void* d_out, int out_size, void* d_ws, size_t ws_size,
                              hipStream_t stream)
{
    const float* feats   = (const float*)d_in[0];   // [B,T,K]
    const int*   tags    = (const int*)  d_in[1];   // [B,T]
    const int*   lengths = (const int*)  d_in[2];   // [B]
    const float* trans   = (const float*)d_in[3];   // [K,K]

    float* fwd  = (float*)d_ws;
    float* gold = fwd + CRF_B;
    float* out  = (float*)d_out;

    crf_scan_kernel<<<CRF_B / 16, 32, 0, stream>>>(feats, lengths, trans, fwd);
    crf_gold_kernel<<<CRF_B / 256, 256, 0, stream>>>(feats, tags, lengths, trans, gold);
    crf_reduce_kernel<<<1, 256, 0, stream>>>(fwd, gold, out);
}